// SumKernel_76811195121789
// MI455X (gfx1250) — compile-verified
//
#include <hip/hip_runtime.h>

// ext-vector types matching CDNA5 WMMA f32 16x16x4 operand sizes:
// A (16x4 f32) = 2 VGPRs/lane, B (4x16 f32) = 2 VGPRs/lane, C/D (16x16 f32) = 8 VGPRs/lane.
typedef __attribute__((ext_vector_type(2))) float v2f;
typedef __attribute__((ext_vector_type(8))) float v8f;

#define B_  4
#define P_  64
#define S_  8
#define YD  16
#define GD  48

constexpr int BLOCK = 256;                   // 8 waves per workgroup (wave32)
constexpr int GRID  = 1024;
constexpr int WAVES = GRID * (BLOCK / 32);   // 8192 waves -> stride is compile-time

// ---------------------------------------------------------------------------
// Kernel 1: tiny precompute of the broadcast tables (8 KB each, L2 resident).
//   ky1[b,p,s] = dot(y[b,p,s,:], Wy[0:16])  + by + bg   (both scalars folded)
//   ky2[b,p,s] = dot(y[b,p,s,:], Wy[16:32])
// ---------------------------------------------------------------------------
__global__ void ky_tables_kernel(const float* __restrict__ y,
                                 const float* __restrict__ Wy,
                                 const float* __restrict__ by,
                                 const float* __restrict__ bg,
                                 float* __restrict__ ky1,
                                 float* __restrict__ ky2) {
    int i = blockIdx.x * blockDim.x + threadIdx.x;
    if (i >= B_ * P_ * S_) return;
    const float* yr = y + (size_t)i * YD;
    float d1 = 0.f, d2 = 0.f;
#pragma unroll
    for (int j = 0; j < YD; ++j) {
        float v = yr[j];
        d1 += v * Wy[j];
        d2 += v * Wy[YD + j];
    }
    ky1[i] = d1 + by[0] + bg[0];
    ky2[i] = d2;
}

// ---------------------------------------------------------------------------
// Kernel 2: HBM-bound GEMV via V_WMMA_F32_16X16X4_F32, two tiles per wave
// iteration.
//   A[m][k] = Wg[kbase+k]  (rows replicated -> D independent of m)
//   B[k][n] = g[(tile*16+n)*48 + kbase + k]
// Each of the two tiles runs its own independent 12-deep WMMA accumulation
// chain (K=48 in chunks of 4) from a zero accumulator; lane L ends up holding
// tile0's row L (L<16) or tile1's row L-16 (L>=16) in c[0], so the epilogue
// is a single fully-coalesced unmasked 128B store.
// ---------------------------------------------------------------------------
__global__ void __launch_bounds__(BLOCK)
kg_wmma_kernel(const float* __restrict__ g,
               const float* __restrict__ Wg,
               const float* __restrict__ ky1,
               const float* __restrict__ ky2,
               float* __restrict__ out,
               int npairs) {
    const int lane = threadIdx.x & 31;
    const int wave = blockIdx.x * (BLOCK >> 5) + (threadIdx.x >> 5);
    const int n  = lane & 15;   // tile column == output row within tile
    const int kh = lane >> 4;   // which K-pair this lane supplies (0 or 1)

    // Preload A fragments: aw[j] = (Wg[4j + 2kh], Wg[4j + 2kh + 1]).
    v2f aw[12];
#pragma unroll
    for (int j = 0; j < 12; ++j) {
        aw[j].x = Wg[4 * j + 2 * kh];
        aw[j].y = Wg[4 * j + 2 * kh + 1];
    }

    for (int p = wave; p < npairs; p += WAVES) {
        const long base = (long)p * 32;          // 32 outputs per pair of tiles
        const long oe   = base + lane;           // this lane's output element

        // Broadcast term for THIS lane's output (stride 2^18 across iterations
        // -> s1,s2,p1,p2 are loop-invariant, only b varies; compiler hoists).
        const int s2 = (int)(oe & 7);
        const int s1 = (int)((oe >> 3) & 7);
        const int p2 = (int)((oe >> 6) & 63);
        const int p1 = (int)((oe >> 12) & 63);
        const int b  = (int)(oe >> 18);
        const float e = ky1[((b * P_ + p2) << 3) + s2]
                      + ky2[((b * P_ + p1) << 3) + s1];

        // Streaming reads: each lane reads its K-half of g-row n for both
        // tiles; 24 NT b64 loads = 6 KB per wave-iteration, each byte once.
        const float* g0 = g + (size_t)(base + n) * GD + 2 * kh;   // tile p*2
        const float* g1 = g0 + 16 * GD;                           // tile p*2+1

        v8f c0 = {0.f, 0.f, 0.f, 0.f, 0.f, 0.f, 0.f, 0.f};
        v8f c1 = {0.f, 0.f, 0.f, 0.f, 0.f, 0.f, 0.f, 0.f};
#pragma unroll
        for (int j = 0; j < 12; ++j) {
            v2f bv0 = __builtin_nontemporal_load((const v2f*)(g0 + 4 * j));
            v2f bv1 = __builtin_nontemporal_load((const v2f*)(g1 + 4 * j));
            // 8 args: (neg_a, A, neg_b, B, c_mod, C, reuse_a, reuse_b)
            c0 = __builtin_amdgcn_wmma_f32_16x16x4_f32(
                     false, aw[j], false, bv0, (short)0, c0, false, false);
            c1 = __builtin_amdgcn_wmma_f32_16x16x4_f32(
                     false, aw[j], false, bv1, (short)0, c1, false, false);
        }

        // Lanes 0-15 hold tile0 rows, lanes 16-31 hold tile1 rows: one
        // contiguous 128B NT store from the full wave, no exec masking.
        const float val = (lane < 16 ? c0[0] : c1[0]) + e;
        __builtin_nontemporal_store(val, out + oe);
    }
}

// ---------------------------------------------------------------------------
// Inputs (setup_inputs order): 0:y 1:pairwise_g 2:Wy 3:by 4:Wg 5:bg
// Output: float32, B*P*P*S*S = 1,048,576 elements.
// ---------------------------------------------------------------------------
extern "C" void kernel_launch(void* const* d_in, const int* in_sizes, int n_in,
                              void* d_out, int out_size, void* d_ws, size_t ws_size,
                              hipStream_t stream) {
    const float* y  = (const float*)d_in[0];
    const float* g  = (const float*)d_in[1];
    const float* Wy = (const float*)d_in[2];
    const float* by = (const float*)d_in[3];
    const float* Wg = (const float*)d_in[4];
    const float* bg = (const float*)d_in[5];
    float* out = (float*)d_out;

    float* ky1 = (float*)d_ws;            // 2048 floats
    float* ky2 = ky1 + (B_ * P_ * S_);    // 2048 floats (16 KB total)

    ky_tables_kernel<<<(B_ * P_ * S_ + 255) / 256, 256, 0, stream>>>(
        y, Wy, by, bg, ky1, ky2);

    const int npairs = out_size / 32;     // 32,768 tile-pairs -> 4 per wave
    kg_wmma_kernel<<<GRID, BLOCK, 0, stream>>>(g, Wg, ky1, ky2, out, npairs);
}